// GraphConvolution_2465311228496
// MI455X (gfx1250) — compile-verified
//
#include <hip/hip_runtime.h>
#include <hip/hip_bf16.h>

typedef float v2f __attribute__((ext_vector_type(2)));
typedef float v8f __attribute__((ext_vector_type(8)));

#define D_IN  128
#define D_OUT 128

// ---------------------------------------------------------------------------
// Kernel 1: pre_sup[s] = x @ W[s]   via V_WMMA_F32_16X16X4_F32
//   block = 256 threads (8 waves); each wave -> 16 rows x 128 cols
//   grid  = (ceil(N/128), 2 supports)
//   W[s] (64 KB) transposed into LDS so B-fragments are ds_load_b64.
// ---------------------------------------------------------------------------
__global__ __launch_bounds__(256) void gcn_gemm_wmma(
    const float* __restrict__ x,      // [N, 128]
    const float* __restrict__ W,      // [2, 128, 128]
    float* __restrict__ pre,          // [2, N, 128]
    int nNodes)
{
    __shared__ float wt[D_OUT * D_IN];   // wt[n*128 + k] = W[s][k][n]  (64 KB)

    const int s = blockIdx.y;
    const float* Ws = W + (size_t)s * D_IN * D_OUT;

    for (int i = threadIdx.x; i < D_IN * D_OUT; i += 256) {
        const int k = i >> 7;
        const int n = i & 127;
        wt[n * D_IN + k] = Ws[i];
    }
    __syncthreads();

    const int wave    = threadIdx.x >> 5;
    const int lane    = threadIdx.x & 31;
    const int rowBase = blockIdx.x * 128 + wave * 16;

    const int mIdx = lane & 15;
    const int kOff = (lane >> 4) << 1;

    int arow = rowBase + mIdx;
    if (arow >= nNodes) arow = nNodes - 1;
    const float* aptr = x + (size_t)arow * D_IN + kOff;

    v8f c[8] = {};

    for (int k0 = 0; k0 < D_IN; k0 += 4) {
        const v2f a = *(const v2f*)(aptr + k0);
#pragma unroll
        for (int nt = 0; nt < 8; ++nt) {
            const int n = nt * 16 + mIdx;
            const v2f b = *(const v2f*)(&wt[n * D_IN + k0 + kOff]);
            c[nt] = __builtin_amdgcn_wmma_f32_16x16x4_f32(
                        false, a, false, b, (short)0, c[nt], false, false);
        }
    }

    float* outBase = pre + (size_t)s * nNodes * D_OUT;
    const int rowHalf = (lane >> 4) << 3;
#pragma unroll
    for (int v = 0; v < 8; ++v) {
        const int row = rowBase + rowHalf + v;
        if (row < nNodes) {
#pragma unroll
            for (int nt = 0; nt < 8; ++nt) {
                outBase[(size_t)row * D_OUT + nt * 16 + mIdx] = c[nt][v];
            }
        }
    }
}

// ---------------------------------------------------------------------------
// CSR construction: zero counts -> count -> scan -> cursors -> scatter
// ---------------------------------------------------------------------------
__global__ __launch_bounds__(256) void gcn_zero_i32(int* __restrict__ p, int n)
{
    const int i = blockIdx.x * 256 + threadIdx.x;
    if (i < n) p[i] = 0;
}

__global__ __launch_bounds__(256) void gcn_count(
    const int* __restrict__ ei0, const int* __restrict__ ei1,
    int* __restrict__ cnt, int nNodes, int nE)
{
    const int t = blockIdx.x * 256 + threadIdx.x;
    if (t >= nE) return;
    atomicAdd(&cnt[ei0[t]], 1);            // support 0 rows
    atomicAdd(&cnt[nNodes + ei1[t]], 1);   // support 1 rows
}

// single-block chunked exclusive scan over cnt[s][0..N) -> rowStart[s][0..N]
__global__ __launch_bounds__(1024) void gcn_scan(
    const int* __restrict__ cnt, int* __restrict__ rowStart, int nNodes, int nE)
{
    __shared__ int buf[1024];
    __shared__ int carry;
    for (int s = 0; s < 2; ++s) {
        if (threadIdx.x == 0) carry = 0;
        __syncthreads();
        for (int base = 0; base < nNodes; base += 1024) {
            const int i = base + threadIdx.x;
            const int v = (i < nNodes) ? cnt[s * nNodes + i] : 0;
            buf[threadIdx.x] = v;
            __syncthreads();
            for (int d = 1; d < 1024; d <<= 1) {
                const int t = (threadIdx.x >= d) ? buf[threadIdx.x - d] : 0;
                __syncthreads();
                buf[threadIdx.x] += t;
                __syncthreads();
            }
            const int incl = buf[threadIdx.x];
            if (i < nNodes) rowStart[s * (nNodes + 1) + i] = carry + incl - v;
            __syncthreads();
            if (threadIdx.x == 1023) carry += incl;   // chunk total
            __syncthreads();
        }
        if (threadIdx.x == 0) rowStart[s * (nNodes + 1) + nNodes] = nE;
        __syncthreads();
    }
}

__global__ __launch_bounds__(256) void gcn_init_cursor(
    const int* __restrict__ rowStart, int* __restrict__ cursor, int nNodes)
{
    const int i = blockIdx.x * 256 + threadIdx.x;
    if (i >= 2 * nNodes) return;
    const int s = (i >= nNodes) ? 1 : 0;
    const int r = i - s * nNodes;
    cursor[i] = rowStart[s * (nNodes + 1) + r];
}

__global__ __launch_bounds__(256) void gcn_scatter(
    const int* __restrict__ ei0, const float* __restrict__ v0,
    const int* __restrict__ ei1, const float* __restrict__ v1,
    int* __restrict__ cursor,
    int* __restrict__ sCol, float* __restrict__ sVal,
    int nNodes, int nE)
{
    const int t = blockIdx.x * 256 + threadIdx.x;
    if (t >= nE) return;
    {   // support 0
        const int r = ei0[t];
        const int p = atomicAdd(&cursor[r], 1);
        sCol[p] = ei0[nE + t];
        sVal[p] = v0[t];
    }
    {   // support 1
        const int r = ei1[t];
        const int p = atomicAdd(&cursor[nNodes + r], 1);
        sCol[nE + p] = ei1[nE + t];
        sVal[nE + p] = v1[t];
    }
}

// ---------------------------------------------------------------------------
// Kernel 7: gather-only segmented SpMM + fused ReLU.
//   one wave per output row; lane owns a float4 (32*4 = 128 features);
//   registers accumulate over both supports; single plain store, no atomics.
// ---------------------------------------------------------------------------
__global__ __launch_bounds__(256) void gcn_spmm_csr(
    const int*   __restrict__ rowStart,  // [2][N+1]
    const int*   __restrict__ sCol,      // [2][E]
    const float* __restrict__ sVal,      // [2][E]
    const float* __restrict__ pre,       // [2][N][128]
    float*       __restrict__ out,       // [N][128]
    int nNodes, int nE)
{
    const int row = blockIdx.x * 8 + (threadIdx.x >> 5);
    if (row >= nNodes) return;
    const int lane = threadIdx.x & 31;

    float4 acc = make_float4(0.f, 0.f, 0.f, 0.f);

#pragma unroll
    for (int s = 0; s < 2; ++s) {
        const float4* preS = (const float4*)(pre + (size_t)s * nNodes * D_OUT);
        const int*    cS   = sCol + (size_t)s * nE;
        const float*  vS   = sVal + (size_t)s * nE;
        const int beg = rowStart[s * (nNodes + 1) + row];
        const int end = rowStart[s * (nNodes + 1) + row + 1];
        for (int j = beg; j < end; ++j) {
            const int   c = cS[j];
            const float v = vS[j];
            if (j + 1 < end)   // hide gather latency: global_prefetch_b8
                __builtin_prefetch(&preS[(size_t)cS[j + 1] * 32 + lane], 0, 0);
            const float4 g = preS[(size_t)c * 32 + lane];
            acc.x += v * g.x; acc.y += v * g.y;
            acc.z += v * g.z; acc.w += v * g.w;
        }
    }

    float4 r;
    r.x = fmaxf(acc.x, 0.f); r.y = fmaxf(acc.y, 0.f);
    r.z = fmaxf(acc.z, 0.f); r.w = fmaxf(acc.w, 0.f);
    ((float4*)(out + (size_t)row * D_OUT))[lane] = r;
}

// ---------------------------------------------------------------------------
extern "C" void kernel_launch(void* const* d_in, const int* in_sizes, int n_in,
                              void* d_out, int out_size, void* d_ws, size_t ws_size,
                              hipStream_t stream)
{
    const float* x   = (const float*)d_in[0];
    const float* W   = (const float*)d_in[1];
    const int*   ei0 = (const int*)  d_in[2];
    const float* v0  = (const float*)d_in[3];
    const int*   ei1 = (const int*)  d_in[4];
    const float* v1  = (const float*)d_in[5];

    float* out = (float*)d_out;

    const int nNodes = in_sizes[0] / D_IN;   // 50000
    const int nE     = in_sizes[3];          // 800000

    // workspace layout
    char* wp = (char*)d_ws;
    float* pre      = (float*)wp; wp += (size_t)2 * nNodes * D_OUT * 4;  // 51.2 MB
    int*   cnt      = (int*)wp;   wp += (size_t)2 * nNodes * 4;
    int*   rowStart = (int*)wp;   wp += (size_t)2 * (nNodes + 1) * 4;
    int*   cursor   = (int*)wp;   wp += (size_t)2 * nNodes * 4;
    int*   sCol     = (int*)wp;   wp += (size_t)2 * nE * 4;              // 6.4 MB
    float* sVal     = (float*)wp; wp += (size_t)2 * nE * 4;              // 6.4 MB

    // ---- CSR build ----
    gcn_zero_i32<<<dim3((2 * nNodes + 255) / 256), dim3(256), 0, stream>>>(
        cnt, 2 * nNodes);
    gcn_count<<<dim3((nE + 255) / 256), dim3(256), 0, stream>>>(
        ei0, ei1, cnt, nNodes, nE);
    gcn_scan<<<dim3(1), dim3(1024), 0, stream>>>(cnt, rowStart, nNodes, nE);
    gcn_init_cursor<<<dim3((2 * nNodes + 255) / 256), dim3(256), 0, stream>>>(
        rowStart, cursor, nNodes);
    gcn_scatter<<<dim3((nE + 255) / 256), dim3(256), 0, stream>>>(
        ei0, v0, ei1, v1, cursor, sCol, sVal, nNodes, nE);

    // ---- dense GEMM for both supports (WMMA) ----
    dim3 gg((nNodes + 127) / 128, 2);
    gcn_gemm_wmma<<<gg, dim3(256), 0, stream>>>(x, W, pre, nNodes);

    // ---- gather-only SpMM + fused ReLU ----
    gcn_spmm_csr<<<dim3((nNodes + 7) / 8), dim3(256), 0, stream>>>(
        rowStart, sCol, sVal, pre, out, nNodes, nE);
}